// UserInnerContrastiveLoss_50611894616416
// MI455X (gfx1250) — compile-verified
//
#include <hip/hip_runtime.h>
#include <hip/hip_bf16.h>

typedef __attribute__((ext_vector_type(2))) float v2f;
typedef __attribute__((ext_vector_type(8))) float v8f;

#define NROWS 16384
#define NCOLS 512
#define INV_T 10.0f

// Extract element i (0..7) from a v8f with a small select chain (i is
// uniform-ish per lane; 7 v_cndmask ops, negligible).
__device__ __forceinline__ float pick8(v8f v, int i) {
    float r = v[0];
#pragma unroll
    for (int j = 1; j < 8; ++j) r = (i == j) ? v[j] : r;
    return r;
}

// One wave handles 16 rows. Computes the four 16x16 gram tiles
// (si*sj^T, si*pj^T, pi*sj^T, pi*pj^T) with V_WMMA_F32_16X16X4_F32
// accumulation over K=512, extracts the diagonals (per-row dot products),
// and writes the per-row CE loss term to row_loss.
__global__ __launch_bounds__(256) void contrastive_dots_wmma(
    const float* __restrict__ si, const float* __restrict__ pi,
    const float* __restrict__ sj, const float* __restrict__ pj,
    float* __restrict__ row_loss)
{
    const int lane  = threadIdx.x & 31;
    const int wave  = threadIdx.x >> 5;
    const int group = blockIdx.x * 8 + wave;   // 16-row group, 0..1023
    const int row0  = group * 16;

    // A-tile (16x4 f32): lane L holds row L%16, K = 2*(L>>4)+{0,1}.
    // B-tile (4x16 f32): lane L holds col L%16 (a j-row), same K split.
    const int  m = lane & 15;
    const int  h = lane >> 4;
    const long base = (long)(row0 + m) * NCOLS + 2 * h;

    const float* __restrict__ as_p = si + base;
    const float* __restrict__ ap_p = pi + base;
    const float* __restrict__ bs_p = sj + base;
    const float* __restrict__ bp_p = pj + base;

    v8f c_pos = {}; v8f c_n1 = {}; v8f c_n2 = {}; v8f c_n3 = {};

#pragma unroll 4
    for (int k = 0; k < NCOLS; k += 4) {
        v2f a_s = *(const v2f*)(as_p + k);
        v2f a_p = *(const v2f*)(ap_p + k);
        v2f b_s = *(const v2f*)(bs_p + k);
        v2f b_p = *(const v2f*)(bp_p + k);
        // D = A*B + C ; each loaded operand feeds two WMMAs.
        c_pos = __builtin_amdgcn_wmma_f32_16x16x4_f32(
            false, a_s, false, b_s, (short)0, c_pos, false, false);
        c_n1  = __builtin_amdgcn_wmma_f32_16x16x4_f32(
            false, a_s, false, b_p, (short)0, c_n1, false, false);
        c_n2  = __builtin_amdgcn_wmma_f32_16x16x4_f32(
            false, a_p, false, b_s, (short)0, c_n2, false, false);
        c_n3  = __builtin_amdgcn_wmma_f32_16x16x4_f32(
            false, a_p, false, b_p, (short)0, c_n3, false, false);
    }

    // C/D layout: VGPR v, lanes 0-15 -> (M=v,   N=lane),
    //                     lanes16-31 -> (M=v+8, N=lane-16).
    // Diag m<8 : lane m,    vgpr m.   Diag m>=8: lane m+16, vgpr m-8.
    const bool lo = (lane < 8);
    const bool hi = (lane >= 24);
    if (lo || hi) {
        const int idx = lo ? lane : (lane - 24);   // vgpr index 0..7
        const int row = lo ? lane : (lane - 16);   // row in group 0..15

        float l0 = INV_T * pick8(c_pos, idx);
        float l1 = INV_T * pick8(c_n1,  idx);
        float l2 = INV_T * pick8(c_n2,  idx);
        float l3 = INV_T * pick8(c_n3,  idx);

        float mx = fmaxf(fmaxf(l0, l1), fmaxf(l2, l3));
        float s  = expf(l0 - mx) + expf(l1 - mx) + expf(l2 - mx) + expf(l3 - mx);
        // -log_softmax[0] = logsumexp(logits) - logits[0]
        row_loss[row0 + row] = (mx + logf(s)) - l0;
    }
}

// Single-block final reduction: mean over NROWS per-row losses.
__global__ __launch_bounds__(256) void contrastive_reduce(
    const float* __restrict__ row_loss, float* __restrict__ out)
{
    __shared__ float sm[256];
    float s = 0.0f;
    for (int i = threadIdx.x; i < NROWS; i += 256) s += row_loss[i];
    sm[threadIdx.x] = s;
    __syncthreads();
#pragma unroll
    for (int off = 128; off > 0; off >>= 1) {
        if ((int)threadIdx.x < off) sm[threadIdx.x] += sm[threadIdx.x + off];
        __syncthreads();
    }
    if (threadIdx.x == 0) out[0] = sm[0] * (1.0f / (float)NROWS);
}

extern "C" void kernel_launch(void* const* d_in, const int* in_sizes, int n_in,
                              void* d_out, int out_size, void* d_ws, size_t ws_size,
                              hipStream_t stream) {
    const float* si = (const float*)d_in[0];  // shared_i   [16384,512]
    const float* pi = (const float*)d_in[1];  // specific_i
    const float* sj = (const float*)d_in[2];  // shared_j
    const float* pj = (const float*)d_in[3];  // specific_j
    float* out      = (float*)d_out;
    float* row_loss = (float*)d_ws;           // NROWS floats (64 KB) scratch

    // 1024 groups of 16 rows; 8 waves (256 threads) per block -> 128 blocks.
    contrastive_dots_wmma<<<128, 256, 0, stream>>>(si, pi, sj, pj, row_loss);
    contrastive_reduce<<<1, 256, 0, stream>>>(row_loss, out);
}